// GIN_46033459478999
// MI455X (gfx1250) — compile-verified
//
#include <hip/hip_runtime.h>
#include <math.h>

#define NNODES 50000
#define MPAD   50048          // 782 * 64 : row-padded workspace leading dim
#define NEDGES 800000
#define IND    128
#define HD     256
#define CATD   768
#define BN_EPS 1e-5f

#define BM 64                 // block output tile rows
#define BN 64                 // block output tile cols
#define KC 32                 // K chunk staged in LDS
#define LDAT 36               // padded LDS stride for A tile (mult of 4, bank-spread)

typedef float v2f __attribute__((ext_vector_type(2)));
typedef float v8f __attribute__((ext_vector_type(8)));
typedef __attribute__((address_space(3))) float* lds_fp;

// ------------------------------------------------ CDNA5 async copy primitives
__device__ __forceinline__ void async_b128(float* lds_dst, const float* gsrc) {
  lds_fp l = (lds_fp)lds_dst;   // generic -> LDS addrspace (32-bit offset VGPR)
  asm volatile("global_load_async_to_lds_b128 %0, %1, off"
               :: "v"(l), "v"(gsrc) : "memory");
}
__device__ __forceinline__ void wait_async_le(int) {}
__device__ __forceinline__ void wait_async3() {
  asm volatile("s_wait_asynccnt 0x3" ::: "memory");
}
__device__ __forceinline__ void wait_async0() {
  asm volatile("s_wait_asynccnt 0x0" ::: "memory");
}

// ---------------------------------------------------------------- utilities
__global__ void k_zero(float* __restrict__ p, int n) {
  int t = blockIdx.x * blockDim.x + threadIdx.x;
  if (t < n) p[t] = 0.f;
}

__global__ void k_copy_strided(float* __restrict__ dst, int lddst,
                               const float* __restrict__ src, int ldsrc,
                               int n, int d) {
  int t = blockIdx.x * blockDim.x + threadIdx.x;
  int chunks = d >> 2;
  if (t >= n * chunks) return;
  int row = t / chunks;
  int c   = (t % chunks) << 2;
  float4 v = *(const float4*)(src + (size_t)row * ldsrc + c);
  *(float4*)(dst + (size_t)row * lddst + c) = v;
}

// t[dst[e], :] += x[src[e], :]
__global__ void k_scatter_add(float* __restrict__ t, int d,
                              const float* __restrict__ x, int ldx,
                              const int* __restrict__ ei, int ne) {
  int id = blockIdx.x * blockDim.x + threadIdx.x;
  int chunks = d >> 2;
  if (id >= ne * chunks) return;
  int e = id / chunks;
  int c = (id % chunks) << 2;
  int s  = ei[e];
  int ds = ei[ne + e];
  float4 v = *(const float4*)(x + (size_t)s * ldx + c);
  float* tp = t + (size_t)ds * d + c;
  unsafeAtomicAdd(tp + 0, v.x);
  unsafeAtomicAdd(tp + 1, v.y);
  unsafeAtomicAdd(tp + 2, v.z);
  unsafeAtomicAdd(tp + 3, v.w);
}

// ----------------------------------------------- LDS-tiled fp32 WMMA GEMM
// C[MPAD,Ncol] = A[MPAD,K] @ B[K,Ncol] + bias  (optionally ReLU)
// Block: 256 threads (8 waves) -> 64x64 output tile; wave -> 16x32 via two
// V_WMMA_F32_16X16X4_F32 accumulators. A/B K-chunks staged in LDS with
// double-buffered global_load_async_to_lds_b128 (ASYNCcnt) overlapped with
// the WMMA compute on the other buffer.
template <bool RELU>
__global__ void __launch_bounds__(256)
k_gemm_wmma(const float* __restrict__ A, int lda,
            const float* __restrict__ B, int ldb,
            const float* __restrict__ bias,
            float* __restrict__ C, int ldc,
            int K, int Ncol) {
  __shared__ float lA[2][BM * LDAT];   // [64][36] padded
  __shared__ float lB[2][KC * BN];     // [32][64]

  const int tid  = threadIdx.x;
  const int wave = tid >> 5;
  const int lane = tid & 31;
  const int wr   = wave >> 1;          // 0..3 row group
  const int wc   = wave & 1;           // 0..1 col group
  const int half = lane >> 4;          // K pair select
  const int idx  = lane & 15;

  const int m0 = blockIdx.y * BM;
  const int n0 = blockIdx.x * BN;

  // loader mapping (256 threads fill B and A rows 0..31, 3 async b128 per thread)
  const int ar = tid >> 3, ac = (tid & 7) << 2;    // A: 32 rows x 8 chunks
  const int br = tid >> 4, bc = (tid & 15) << 2;   // B: 16 rows x 16 chunks (x2)

  const float* Ag = A + (size_t)(m0 + ar) * lda + ac;
  const float* Bg0 = B + (size_t)br * ldb + n0 + bc;
  const float* Bg1 = B + (size_t)(br + 16) * ldb + n0 + bc;

  // prologue: stage chunk 0 into buffer 0
  async_b128(&lA[0][ar * LDAT + ac], Ag);
  async_b128(&lB[0][br * BN + bc], Bg0);
  async_b128(&lB[0][(br + 16) * BN + bc], Bg1);

  v8f acc0 = {};
  v8f acc1 = {};

  const int rbase = wr * 16;
  const int cbase = wc * 32;
  const int nchunk = K / KC;

  for (int ch = 0; ch < nchunk; ++ch) {
    const int cur = ch & 1;
    // kick off next chunk's DMA into the other buffer
    if (ch + 1 < nchunk) {
      const int nxt = cur ^ 1;
      const size_t koff = (size_t)(ch + 1) * KC;
      async_b128(&lA[nxt][ar * LDAT + ac], Ag + koff);
      async_b128(&lB[nxt][br * BN + bc], Bg0 + koff * ldb);
      async_b128(&lB[nxt][(br + 16) * BN + bc], Bg1 + koff * ldb);
      wait_async3();                    // own current-chunk transfers done
    } else {
      wait_async0();
    }
    __syncthreads();                    // everyone's current chunk visible

    const float* Ab = &lA[cur][(rbase + idx) * LDAT + 2 * half];
    const float* Bb = &lB[cur][(2 * half) * BN + cbase + idx];
#pragma unroll
    for (int k = 0; k < KC; k += 4) {
      v2f a;  a.x  = Ab[k];            a.y  = Ab[k + 1];
      v2f b0; b0.x = Bb[k * BN];       b0.y = Bb[k * BN + BN];
      v2f b1; b1.x = Bb[k * BN + 16];  b1.y = Bb[k * BN + BN + 16];
      acc0 = __builtin_amdgcn_wmma_f32_16x16x4_f32(false, a, false, b0,
                                                   (short)0, acc0, false, false);
      acc1 = __builtin_amdgcn_wmma_f32_16x16x4_f32(false, a, false, b1,
                                                   (short)0, acc1, false, false);
    }
    __syncthreads();                    // done reading cur before it's refilled
  }

  const float bi0 = bias[n0 + cbase + idx];
  const float bi1 = bias[n0 + cbase + 16 + idx];
#pragma unroll
  for (int j = 0; j < 8; ++j) {
    int row = m0 + rbase + j + 8 * half;       // C layout: vgpr j -> row j / j+8
    float c0 = acc0[j] + bi0;
    float c1 = acc1[j] + bi1;
    if (RELU) { c0 = fmaxf(c0, 0.f); c1 = fmaxf(c1, 0.f); }
    C[(size_t)row * ldc + n0 + cbase + idx]      = c0;
    C[(size_t)row * ldc + n0 + cbase + 16 + idx] = c1;
  }
}

// -------------------------------------------- batch-norm stats + apply+ReLU
__global__ void k_colstats(const float* __restrict__ h, int n, int d,
                           float* __restrict__ sums, float* __restrict__ sqs) {
  int col = threadIdx.x;
  int rowsPer = (n + gridDim.x - 1) / gridDim.x;
  int r0 = blockIdx.x * rowsPer;
  int r1 = min(n, r0 + rowsPer);
  float s = 0.f, q = 0.f;
  for (int r = r0; r < r1; ++r) {
    float v = h[(size_t)r * d + col];
    s += v;
    q += v * v;
  }
  unsafeAtomicAdd(&sums[col], s);
  unsafeAtomicAdd(&sqs[col], q);
}

__global__ void k_bn_relu(float* __restrict__ h, int n, int d,
                          const float* __restrict__ sums,
                          const float* __restrict__ sqs,
                          const float* __restrict__ g,
                          const float* __restrict__ be) {
  int t = blockIdx.x * blockDim.x + threadIdx.x;
  if (t >= n * d) return;
  int col = t % d;
  float invn = 1.f / (float)n;
  float mu   = sums[col] * invn;
  float var  = sqs[col] * invn - mu * mu;
  float v = (h[t] - mu) * rsqrtf(var + BN_EPS) * g[col] + be[col];
  h[t] = fmaxf(v, 0.f);
}

// ---------------------- head: logits = r @ W[768,2] + b, softmax, both halves
__global__ void k_head(const float* __restrict__ r,
                       const float* __restrict__ W,
                       const float* __restrict__ b,
                       float* __restrict__ out, int n) {
  int gid  = blockIdx.x * blockDim.x + threadIdx.x;
  int wave = gid >> 5;
  int lane = gid & 31;
  if (wave >= n) return;
  const float* row = r + (size_t)wave * CATD;
  float s0 = 0.f, s1 = 0.f;
  for (int k = lane; k < CATD; k += 32) {
    float v = row[k];
    s0 += v * W[2 * k];
    s1 += v * W[2 * k + 1];
  }
  for (int off = 16; off; off >>= 1) {
    s0 += __shfl_xor(s0, off, 32);
    s1 += __shfl_xor(s1, off, 32);
  }
  if (lane == 0) {
    float l0 = s0 + b[0], l1 = s1 + b[1];
    out[2 * wave]     = l0;
    out[2 * wave + 1] = l1;
    float m  = fmaxf(l0, l1);
    float e0 = __expf(l0 - m), e1 = __expf(l1 - m);
    float inv = 1.f / (e0 + e1);
    out[2 * n + 2 * wave]     = e0 * inv;
    out[2 * n + 2 * wave + 1] = e1 * inv;
  }
}

// ---------------------------------------------------------------- launcher
extern "C" void kernel_launch(void* const* d_in, const int* in_sizes, int n_in,
                              void* d_out, int out_size, void* d_ws, size_t ws_size,
                              hipStream_t stream) {
  const float* x  = (const float*)d_in[0];
  const int*   ei = (const int*)d_in[1];
  const float* c1_W1 = (const float*)d_in[2];
  const float* c1_b1 = (const float*)d_in[3];
  const float* c1_g  = (const float*)d_in[4];
  const float* c1_be = (const float*)d_in[5];
  const float* c1_W2 = (const float*)d_in[6];
  const float* c1_b2 = (const float*)d_in[7];
  const float* c2_W1 = (const float*)d_in[8];
  const float* c2_b1 = (const float*)d_in[9];
  const float* c2_g  = (const float*)d_in[10];
  const float* c2_be = (const float*)d_in[11];
  const float* c2_W2 = (const float*)d_in[12];
  const float* c2_b2 = (const float*)d_in[13];
  const float* c3_W1 = (const float*)d_in[14];
  const float* c3_b1 = (const float*)d_in[15];
  const float* c3_g  = (const float*)d_in[16];
  const float* c3_be = (const float*)d_in[17];
  const float* c3_W2 = (const float*)d_in[18];
  const float* c3_b2 = (const float*)d_in[19];
  const float* lin1_W = (const float*)d_in[20];
  const float* lin1_b = (const float*)d_in[21];
  const float* lin2_W = (const float*)d_in[22];
  const float* lin2_b = (const float*)d_in[23];
  float* out = (float*)d_out;

  // workspace (all matrices row-padded to MPAD rows; pad rows hold garbage
  // that only ever feeds discarded pad outputs)
  float* HCAT  = (float*)d_ws;                       // [MPAD,768] h1|h2|h3
  float* R     = HCAT + (size_t)MPAD * CATD;         // [MPAD,768] readout/scratch
  float* T     = R;                                  // [MPAD,<=256]
  float* HT    = R + (size_t)MPAD * HD;              // [MPAD,256]
  float* STATS = R + (size_t)MPAD * CATD;            // 512 floats
  float* SUMS  = STATS;
  float* SQS   = STATS + HD;

  auto run_conv = [&](const float* src, int ldin, int din,
                      const float* W1, const float* b1,
                      const float* g, const float* be,
                      const float* W2, const float* b2,
                      float* outp, int ldout) {
    int chunks = din >> 2;
    int tot = NNODES * chunks;
    k_copy_strided<<<(tot + 255) / 256, 256, 0, stream>>>(T, din, src, ldin,
                                                          NNODES, din);
    int etot = NEDGES * chunks;
    k_scatter_add<<<(etot + 255) / 256, 256, 0, stream>>>(T, din, src, ldin,
                                                          ei, NEDGES);
    dim3 g1(HD / BN, MPAD / BM);
    k_gemm_wmma<false><<<g1, 256, 0, stream>>>(T, din, W1, HD, b1,
                                               HT, HD, din, HD);
    k_zero<<<2, 256, 0, stream>>>(STATS, 2 * HD);
    k_colstats<<<256, HD, 0, stream>>>(HT, NNODES, HD, SUMS, SQS);
    int tot2 = NNODES * HD;
    k_bn_relu<<<(tot2 + 255) / 256, 256, 0, stream>>>(HT, NNODES, HD,
                                                      SUMS, SQS, g, be);
    k_gemm_wmma<true><<<g1, 256, 0, stream>>>(HT, HD, W2, HD, b2,
                                              outp, ldout, HD, HD);
  };

  run_conv(x, IND, IND, c1_W1, c1_b1, c1_g, c1_be, c1_W2, c1_b2,
           HCAT + 0, CATD);
  run_conv(HCAT + 0, CATD, HD, c2_W1, c2_b1, c2_g, c2_be, c2_W2, c2_b2,
           HCAT + HD, CATD);
  run_conv(HCAT + HD, CATD, HD, c3_W1, c3_b1, c3_g, c3_be, c3_W2, c3_b2,
           HCAT + 2 * HD, CATD);

  // readout: R = relu(HCAT @ lin1_W + lin1_b)   [MPAD,768]x[768,768]
  dim3 gr(CATD / BN, MPAD / BM);
  k_gemm_wmma<true><<<gr, 256, 0, stream>>>(HCAT, CATD, lin1_W, CATD, lin1_b,
                                            R, CATD, CATD, CATD);

  // head + softmax -> d_out (logits then probs)
  k_head<<<(NNODES * 32 + 255) / 256, 256, 0, stream>>>(R, lin2_W, lin2_b,
                                                        out, NNODES);
}